// SpatialSelfAttention_56719338111657
// MI455X (gfx1250) — compile-verified
//
#include <hip/hip_runtime.h>
#include <hip/hip_bf16.h>
#include <math.h>

// Problem constants (match reference)
#define Bq 2
#define Pq 12
#define Nq 325
#define DMq 64
#define Hq 4
#define DKq 16
#define DHq 32
#define Gq 3
#define NODES (Bq * Pq * Nq)          // 7800
#define RPRIME (3 * Hq * DKq)         // 192 rows of meta output per node
#define NT32 ((NODES + 31) / 32)      // 244 blocks of 32 nodes
#define SCALEF 0.25f                  // 1/sqrt(DK)

typedef float v2f __attribute__((ext_vector_type(2)));
typedef float v8f __attribute__((ext_vector_type(8)));

// LDS byte offset from a generic pointer to a __shared__ object:
// low 32 bits of the generic address are the LDS offset (aperture in high bits).
__device__ __forceinline__ unsigned lds_off(const void* p) {
    return (unsigned)(uintptr_t)p;
}

// Per-lane async copy of 16 bytes global -> LDS (ASYNCcnt-tracked).
__device__ __forceinline__ void async_copy_b128(unsigned dst_lds, const void* src) {
    asm volatile("global_load_async_to_lds_b128 %0, %1, off"
                 :: "v"(dst_lds), "v"(src) : "memory");
}

// ---------------------------------------------------------------------------
// Kernel 1: fused meta-learner + per-node meta-weight application -> Q,K,V
//   per block: one branch g, one 32-node tile (2 WMMA M-tiles sharing B)
//   h = relu(c @ l1w^T + l1b)                        [32,32]  (VALU from LDS)
//   for each r' in 0..191 (r' = (qkv,head,k)):
//     E[32,64] = h @ l2w[r'*64 .. r'*64+63, :]^T      (WMMA f32 16x16x4)
//     QKV[n,r'] = sum_m x[n,m] * (E[n,m] + l2b[r'*64+m])
// ---------------------------------------------------------------------------
__global__ __launch_bounds__(128) void qkv_kernel(
    const float* __restrict__ x, const float* __restrict__ c,
    const float* __restrict__ l1w, const float* __restrict__ l1b,
    const float* __restrict__ l2w, const float* __restrict__ l2b,
    float* __restrict__ Qw, float* __restrict__ Kw, float* __restrict__ Vw)
{
    __shared__ float x_lds[32][DMq];        // 8 KB
    __shared__ float c_lds[32][DMq];        // 8 KB
    __shared__ float h_lds[32][DHq];        // 4 KB
    __shared__ float stage[4][32][DMq];     // 32 KB (per-wave staging)

    const int tile = blockIdx.x;
    const int g    = blockIdx.y;
    const int tid  = threadIdx.x;
    const int lane = tid & 31;
    const int wid  = tid >> 5;
    const int t0   = tile * 32;

    const float* l1wg = l1w + (size_t)g * DHq * DMq;
    const float* l1bg = l1b + (size_t)g * DHq;
    const float* l2wg = l2w + (size_t)g * RPRIME * DMq * DHq;
    const float* l2bg = l2b + (size_t)g * RPRIME * DMq;

    // Async-stage x and c tiles into LDS, 16B per lane-op (zero-pad past NODES).
    for (int i = tid; i < 32 * 16; i += 128) {
        int j  = i >> 4;          // node row 0..31
        int m4 = (i & 15) * 4;    // float column
        int t  = t0 + j;
        if (t < NODES) {
            async_copy_b128(lds_off(&x_lds[j][m4]), x + (size_t)t * DMq + m4);
            async_copy_b128(lds_off(&c_lds[j][m4]), c + (size_t)t * DMq + m4);
        } else {
            float4 z = make_float4(0.f, 0.f, 0.f, 0.f);
            *(float4*)&x_lds[j][m4] = z;
            *(float4*)&c_lds[j][m4] = z;
        }
    }
    asm volatile("s_wait_asynccnt 0" ::: "memory");
    __syncthreads();

    // h = relu(c @ l1w^T + l1b)  (1024 outputs / 128 threads)
    for (int i = tid; i < 32 * DHq; i += 128) {
        int j = i >> 5, d = i & 31;
        float acc = l1bg[d];
        for (int m = 0; m < DMq; ++m) acc += c_lds[j][m] * l1wg[d * DMq + m];
        h_lds[j][d] = fmaxf(acc, 0.f);
    }
    __syncthreads();

    // A-fragments of h for both 16-node sub-tiles; reused for every r'.
    // f32 16x16x4 A layout: lanes 0-15 hold A[M=lane, K=k0..k0+1],
    // lanes 16-31 hold A[M=lane-16, K=k0+2..k0+3].
    const int col  = lane & 15;
    const int half = lane >> 4;
    v2f afrag[2][8];
#pragma unroll
    for (int u = 0; u < 2; ++u)
#pragma unroll
        for (int ks = 0; ks < 8; ++ks) {
            afrag[u][ks].x = h_lds[u * 16 + col][ks * 4 + half * 2 + 0];
            afrag[u][ks].y = h_lds[u * 16 + col][ks * 4 + half * 2 + 1];
        }

    // per-lane node -> (b,p,n) for the scattered Q/K/V stores (lanes 0-15)
    int t_me[2], b_me[2], p_me[2], n_me[2];
#pragma unroll
    for (int u = 0; u < 2; ++u) {
        t_me[u] = t0 + u * 16 + col;
        b_me[u] = p_me[u] = n_me[u] = 0;
        if (t_me[u] < NODES) {
            b_me[u] = t_me[u] / (Pq * Nq);
            int rr  = t_me[u] % (Pq * Nq);
            p_me[u] = rr / Nq;
            n_me[u] = rr % Nq;
        }
    }

    for (int r = wid; r < RPRIME; r += 4) {
        // prefetch next r' chunk of l2w into cache hierarchy
        __builtin_prefetch(l2wg + ((size_t)(r + 4) * DMq + col) * DHq, 0, 0);

#pragma unroll
        for (int mt = 0; mt < 4; ++mt) {
            v8f acc0 = {};
            v8f acc1 = {};
            // B layout (4x16): v0 = B[k0+half*2+0, col], v1 = B[k0+half*2+1, col]
            const size_t rowbase =
                ((size_t)r * DMq + mt * 16 + col) * DHq + half * 2;
#pragma unroll
            for (int ks = 0; ks < 8; ++ks) {
                v2f bfrag;
                bfrag.x = l2wg[rowbase + ks * 4 + 0];
                bfrag.y = l2wg[rowbase + ks * 4 + 1];
                acc0 = __builtin_amdgcn_wmma_f32_16x16x4_f32(
                    false, afrag[0][ks], false, bfrag, (short)0, acc0, false, false);
                acc1 = __builtin_amdgcn_wmma_f32_16x16x4_f32(
                    false, afrag[1][ks], false, bfrag, (short)0, acc1, false, false);
            }
            // C/D layout: VGPR j holds row M=j (lanes 0-15) / M=j+8 (16-31), N=col
#pragma unroll
            for (int j2 = 0; j2 < 8; ++j2) {
                stage[wid][j2 + half * 8][mt * 16 + col]      = acc0[j2];
                stage[wid][16 + j2 + half * 8][mt * 16 + col] = acc1[j2];
            }
        }
        // same-wave LDS RAW across lanes: wait for DS writes, fence compiler
        asm volatile("s_wait_dscnt 0" ::: "memory");

        // QKV[n, r'] = sum_m x[n,m] * (E[n,m] + l2b[r'*64+m]), m split over halves
        const float* l2brow = l2bg + (size_t)r * DMq + half * 32;
        const int qkv = r >> 6;           // 0:Q 1:K 2:V
        const int rem = r & 63;
        const int hh = rem >> 4, kk = rem & 15;
        float* dst = (qkv == 0) ? Qw : (qkv == 1) ? Kw : Vw;
#pragma unroll
        for (int u = 0; u < 2; ++u) {
            float v = 0.f;
#pragma unroll 8
            for (int mm = 0; mm < 32; ++mm) {
                float e = stage[wid][u * 16 + col][half * 32 + mm] + l2brow[mm];
                v += x_lds[u * 16 + col][half * 32 + mm] * e;
            }
            v += __shfl_xor(v, 16, 32);
            if (half == 0 && t_me[u] < NODES) {
                size_t off =
                    (((((size_t)g * Bq + b_me[u]) * Pq + p_me[u]) * Hq + hh) * Nq
                     + n_me[u]) * DKq + kk;
                dst[off] = v;
            }
        }
    }
}

// ---------------------------------------------------------------------------
// Kernel 2: masked attention with "nozero" softmax, per (g,b,p,h)
// ---------------------------------------------------------------------------
__global__ __launch_bounds__(256) void attn_kernel(
    const float* __restrict__ Qw, const float* __restrict__ Kw,
    const float* __restrict__ Vw, const float* __restrict__ tmat,
    const float* __restrict__ ag, float* __restrict__ midw)
{
    int id = blockIdx.x;               // ((g*B + b)*P + p)*H + h
    int hh = id & (Hq - 1); id >>= 2;
    int p  = id % Pq; id /= Pq;
    int b  = id % Bq; id /= Bq;
    int g  = id;

    const size_t base =
        ((((size_t)g * Bq + b) * Pq + p) * Hq + hh) * (size_t)Nq * DKq;
    const float* Qb = Qw + base;
    const float* Kb = Kw + base;
    const float* Vb = Vw + base;
    float* Mb = midw + base;

    const float* tmrow_base = (g < 2)
        ? tmat + (size_t)g * Nq * Nq
        : ag + ((size_t)b * Pq + p) * Nq * Nq;

    for (int n = threadIdx.x; n < Nq; n += 256) {
        float q[DKq];
#pragma unroll
        for (int k = 0; k < DKq; ++k) q[k] = Qb[(size_t)n * DKq + k];
        const float* tr = tmrow_base + (size_t)n * Nq;

        // pass 1: row max over masked scores (masked entries contribute 0.0)
        float maxv = -3.4e38f;
        for (int m = 0; m < Nq; ++m) {
            float tmv = tr[m];
            float s = 0.f;
            if (tmv != 0.f) {
                float d = 0.f;
#pragma unroll
                for (int k = 0; k < DKq; ++k) d += q[k] * Kb[(size_t)m * DKq + k];
                s = d * SCALEF;
            }
            maxv = fmaxf(maxv, s);
        }
        // pass 2: e = exp(s-max) only where s != 0; out = (e*tm) @ V / (sum e + 1e-5)
        float denom = 0.f;
        float acc[DKq];
#pragma unroll
        for (int k = 0; k < DKq; ++k) acc[k] = 0.f;
        for (int m = 0; m < Nq; ++m) {
            float tmv = tr[m];
            if (tmv != 0.f) {
                float d = 0.f;
#pragma unroll
                for (int k = 0; k < DKq; ++k) d += q[k] * Kb[(size_t)m * DKq + k];
                float s = d * SCALEF;
                if (s != 0.f) {
                    float e = __expf(s - maxv);
                    denom += e;
                    float w = e * tmv;
#pragma unroll
                    for (int k = 0; k < DKq; ++k) acc[k] += w * Vb[(size_t)m * DKq + k];
                }
            }
        }
        float inv = 1.f / (denom + 1e-5f);
#pragma unroll
        for (int k = 0; k < DKq; ++k) Mb[(size_t)n * DKq + k] = acc[k] * inv;
    }
}

// ---------------------------------------------------------------------------
// Kernel 3: Gated_Dynamic_Connection over heads (gdc1), per (g, node)
// ---------------------------------------------------------------------------
__global__ __launch_bounds__(64) void gdc1_kernel(
    const float* __restrict__ midw, const float* __restrict__ W1,
    const float* __restrict__ W2, float* __restrict__ bout)
{
    int id = blockIdx.x;
    int t = id % NODES;
    int g = id / NODES;
    int b = t / (Pq * Nq);
    int rr = t % (Pq * Nq);
    int p = rr / Nq;
    int n = rr % Nq;

    __shared__ float md[Hq][DKq];
    int tid = threadIdx.x;
    if (tid < Hq * DKq) {
        int hh = tid >> 4, k = tid & 15;
        size_t off = ((((size_t)g * Bq + b) * Pq + p) * Hq + hh) * (size_t)Nq * DKq
                   + (size_t)n * DKq + k;
        md[hh][k] = midw[off];
    }
    __syncthreads();

    int m = tid;                     // 0..63 output feature
    const float* W1g = W1 + (size_t)g * Hq * DKq * DMq;
    const float* W2g = W2 + (size_t)g * Hq * DKq * DMq;
    float a[Hq], gt[Hq];
    float gmax = -3.4e38f;
#pragma unroll
    for (int hh = 0; hh < Hq; ++hh) {
        float s1 = 0.f, s2 = 0.f;
#pragma unroll
        for (int k = 0; k < DKq; ++k) {
            float v = md[hh][k];
            s1 += v * W1g[(hh * DKq + k) * DMq + m];
            s2 += v * W2g[(hh * DKq + k) * DMq + m];
        }
        a[hh]  = s1;
        gt[hh] = fmaxf(s2, 0.f);
        gmax   = fmaxf(gmax, gt[hh]);
    }
    float den = 0.f;
#pragma unroll
    for (int hh = 0; hh < Hq; ++hh) { gt[hh] = __expf(gt[hh] - gmax); den += gt[hh]; }
    float o = 0.f;
#pragma unroll
    for (int hh = 0; hh < Hq; ++hh) o += a[hh] * gt[hh];
    o /= den;
    bout[((size_t)g * NODES + t) * DMq + m] = o;
}

// ---------------------------------------------------------------------------
// Kernel 4: gdc2 over branches + swish gate + residual + LayerNorm, per node
// ---------------------------------------------------------------------------
__global__ __launch_bounds__(64) void final_kernel(
    const float* __restrict__ x, const float* __restrict__ bout,
    const float* __restrict__ g2W1, const float* __restrict__ g2W2,
    const float* __restrict__ wgw, const float* __restrict__ wgb,
    const float* __restrict__ wow, const float* __restrict__ wob,
    const float* __restrict__ gamma, const float* __restrict__ beta,
    float* __restrict__ out)
{
    int t = blockIdx.x;
    int m = threadIdx.x;
    __shared__ float xr[DMq];
    __shared__ float bo[Gq][DMq];
    __shared__ float sw[DMq];
    __shared__ float zz[DMq];

    xr[m] = x[(size_t)t * DMq + m];
    for (int g = 0; g < Gq; ++g)
        bo[g][m] = bout[((size_t)g * NODES + t) * DMq + m];
    __syncthreads();

    // gdc2 over branch axis
    float a[Gq], gt[Gq];
    float gmax = -3.4e38f;
#pragma unroll
    for (int g = 0; g < Gq; ++g) {
        float s1 = 0.f, s2 = 0.f;
        const float* w1 = g2W1 + (size_t)g * DMq * DMq;
        const float* w2 = g2W2 + (size_t)g * DMq * DMq;
        for (int d = 0; d < DMq; ++d) {
            float v = bo[g][d];
            s1 += v * w1[d * DMq + m];
            s2 += v * w2[d * DMq + m];
        }
        a[g]  = s1;
        gt[g] = fmaxf(s2, 0.f);
        gmax  = fmaxf(gmax, gt[g]);
    }
    float den = 0.f;
#pragma unroll
    for (int g = 0; g < Gq; ++g) { gt[g] = __expf(gt[g] - gmax); den += gt[g]; }
    float o = 0.f;
#pragma unroll
    for (int g = 0; g < Gq; ++g) o += a[g] * gt[g];
    o /= den;

    // swish gate: s = (x @ wg^T + b) * o ; sw = s*sigmoid(s)
    float gl = wgb[m];
    for (int d = 0; d < DMq; ++d) gl += xr[d] * wgw[m * DMq + d];
    float s = gl * o;
    sw[m] = s / (1.f + __expf(-s));
    __syncthreads();

    float y = wob[m];
    for (int d = 0; d < DMq; ++d) y += sw[d] * wow[m * DMq + d];
    float z = y + xr[m];
    zz[m] = z;
    __syncthreads();

    float mu = 0.f;
    for (int d = 0; d < DMq; ++d) mu += zz[d];
    mu *= (1.f / DMq);
    float var = 0.f;
    for (int d = 0; d < DMq; ++d) { float dd = zz[d] - mu; var += dd * dd; }
    var *= (1.f / DMq);
    out[(size_t)t * DMq + m] =
        (z - mu) * __frsqrt_rn(var + 1e-5f) * gamma[m] + beta[m];
}

// ---------------------------------------------------------------------------
extern "C" void kernel_launch(void* const* d_in, const int* in_sizes, int n_in,
                              void* d_out, int out_size, void* d_ws, size_t ws_size,
                              hipStream_t stream) {
    (void)in_sizes; (void)n_in; (void)out_size; (void)ws_size;
    const float* inputs = (const float*)d_in[0];
    const float* c_in   = (const float*)d_in[1];
    const float* tmat   = (const float*)d_in[2];
    const float* agph   = (const float*)d_in[3];
    const float* l1w    = (const float*)d_in[4];
    const float* l1b    = (const float*)d_in[5];
    const float* l2w    = (const float*)d_in[6];
    const float* l2b    = (const float*)d_in[7];
    const float* g1W1   = (const float*)d_in[8];
    const float* g1W2   = (const float*)d_in[9];
    const float* g2W1   = (const float*)d_in[10];
    const float* g2W2   = (const float*)d_in[11];
    const float* wgw    = (const float*)d_in[12];
    const float* wgb    = (const float*)d_in[13];
    const float* wow    = (const float*)d_in[14];
    const float* wob    = (const float*)d_in[15];
    const float* gamma  = (const float*)d_in[16];
    const float* beta   = (const float*)d_in[17];

    float* ws = (float*)d_ws;
    const size_t SEG = (size_t)Gq * Bq * Pq * Hq * Nq * DKq;   // 1,497,600 floats
    float* Qw   = ws;
    float* Kw   = ws + SEG;
    float* Vw   = ws + 2 * SEG;
    float* midw = ws + 3 * SEG;
    float* bout = ws + 4 * SEG;   // G*NODES*DM == SEG floats

    qkv_kernel<<<dim3(NT32, Gq), 128, 0, stream>>>(
        inputs, c_in, l1w, l1b, l2w, l2b, Qw, Kw, Vw);
    attn_kernel<<<Gq * Bq * Pq * Hq, 256, 0, stream>>>(
        Qw, Kw, Vw, tmat, agph, midw);
    gdc1_kernel<<<Gq * NODES, 64, 0, stream>>>(midw, g1W1, g1W2, bout);
    final_kernel<<<NODES, 64, 0, stream>>>(
        inputs, bout, g2W1, g2W2, wgw, wgb, wow, wob, gamma, beta,
        (float*)d_out);
}